// HeteroGraphSAGE_59734405152778
// MI455X (gfx1250) — compile-verified
//
#include <hip/hip_runtime.h>
#include <math.h>

#define N_NODES 100000
#define H_DIM   128
#define E_EDGES 1600000
#define N_LAYERS 3
#define OUT_DIM 32

typedef float v2f __attribute__((ext_vector_type(2)));
typedef float v8f __attribute__((ext_vector_type(8)));

// ---------------------------------------------------------------------------
// Generic fused GEMM: Y[M x Nout] = act(X[M x K] @ W[K x Nout] + bias)
// One wave per 16x16 tile; blockIdx.x = row tile; wave id = column tile.
// Launch with blockDim.x = 32 * (Nout/16) (Nout in {32,64,128}).
// M (=N_NODES) must be a multiple of 16 (100000 = 6250*16).
// ---------------------------------------------------------------------------
__global__ __launch_bounds__(256)
void gemm_bias_act(const float* __restrict__ X, const float* __restrict__ W,
                   const float* __restrict__ bias, float* __restrict__ Y,
                   int K, int Nout, int doRelu) {
    const int lane = threadIdx.x & 31;
    const int wave = threadIdx.x >> 5;
    const int m0   = blockIdx.x * 16;
    const int n0   = wave * 16;
    const int mm   = lane & 15;          // A row within tile == B/D col within tile
    const int hi   = lane >> 4;          // half-wave select
    const int kb   = hi * 2;             // K sub-offset for A/B fragments

    v8f c = {0.f, 0.f, 0.f, 0.f, 0.f, 0.f, 0.f, 0.f};
    const float* xrow = X + (size_t)(m0 + mm) * K;

    for (int k = 0; k < K; k += 4) {
        v2f a, b;
        a.x = xrow[k + kb];
        a.y = xrow[k + kb + 1];
        b.x = W[(size_t)(k + kb)     * Nout + n0 + mm];
        b.y = W[(size_t)(k + kb + 1) * Nout + n0 + mm];
        c = __builtin_amdgcn_wmma_f32_16x16x4_f32(false, a, false, b,
                                                  (short)0, c, false, false);
    }

    const int   col = n0 + mm;
    const float bv  = bias[col];
#pragma unroll
    for (int r = 0; r < 8; ++r) {
        float v = c[r] + bv;
        if (doRelu) v = fmaxf(v, 0.f);
        Y[(size_t)(m0 + r + 8 * hi) * Nout + col] = v;
    }
}

// ---------------------------------------------------------------------------
// Per-destination in-degree (as float, for mean aggregation)
// ---------------------------------------------------------------------------
__global__ __launch_bounds__(256)
void degree_kernel(const int* __restrict__ dst, int E, float* __restrict__ deg) {
    int e = blockIdx.x * blockDim.x + threadIdx.x;
    if (e < E)
        __hip_atomic_fetch_add(&deg[dst[e]], 1.0f,
                               __ATOMIC_RELAXED, __HIP_MEMORY_SCOPE_AGENT);
}

// ---------------------------------------------------------------------------
// Edge scatter-add: agg[dst[e], :] += x[src[e], :].  One wave per edge,
// each lane moves a float4 (32 lanes * 4 = 128 features).
// E must be a multiple of (blockDim/32)*gridDim coverage: 1.6M = 200000*8.
// ---------------------------------------------------------------------------
__global__ __launch_bounds__(256)
void scatter_add_kernel(const float* __restrict__ x, const int* __restrict__ src,
                        const int* __restrict__ dst, int E, float* __restrict__ agg) {
    const int e = blockIdx.x * (blockDim.x >> 5) + (threadIdx.x >> 5);
    if (e >= E) return;
    const int lane = threadIdx.x & 31;
    const int s = src[e];
    const int d = dst[e];
    const float4 v = *(const float4*)(x + (size_t)s * H_DIM + lane * 4);
    float* p = agg + (size_t)d * H_DIM + lane * 4;
    __hip_atomic_fetch_add(p + 0, v.x, __ATOMIC_RELAXED, __HIP_MEMORY_SCOPE_AGENT);
    __hip_atomic_fetch_add(p + 1, v.y, __ATOMIC_RELAXED, __HIP_MEMORY_SCOPE_AGENT);
    __hip_atomic_fetch_add(p + 2, v.z, __ATOMIC_RELAXED, __HIP_MEMORY_SCOPE_AGENT);
    __hip_atomic_fetch_add(p + 3, v.w, __ATOMIC_RELAXED, __HIP_MEMORY_SCOPE_AGENT);
}

// ---------------------------------------------------------------------------
// Fused SAGE layer for one (edge type -> dst node type):
//   out = (agg/max(deg,1)) @ Wl + bl + x_src @ Wr
//   out = out / max(||out||_row, 1e-12)               (L2 row normalize)
//   h_new = relu(BN_eval(out + h_old))
// Block = 256 threads = 8 waves; blockIdx.x = 16-row tile; wave = 16-col tile.
// ---------------------------------------------------------------------------
__global__ __launch_bounds__(256)
void sage_fused_kernel(const float* __restrict__ agg, const float* __restrict__ deg,
                       const float* __restrict__ xsrc, const float* __restrict__ hold,
                       const float* __restrict__ Wl, const float* __restrict__ bl,
                       const float* __restrict__ Wr,
                       const float* __restrict__ bng, const float* __restrict__ bnb,
                       const float* __restrict__ bnm, const float* __restrict__ bnv,
                       float* __restrict__ hnew) {
    __shared__ float sA[16 * H_DIM];      // raw agg tile
    __shared__ float sX[16 * H_DIM];      // x_src tile
    __shared__ float sRD[16];             // 1/max(deg,1) per row
    __shared__ float sPart[8][16];        // per-wave partial row sum-of-squares

    const int t  = threadIdx.x;
    const int m0 = blockIdx.x * 16;

    if (t < 16) sRD[t] = 1.0f / fmaxf(deg[m0 + t], 1.0f);
    for (int i = t; i < 16 * H_DIM; i += 256) {
        const int r = i >> 7, k = i & 127;
        sA[i] = agg [(size_t)(m0 + r) * H_DIM + k];
        sX[i] = xsrc[(size_t)(m0 + r) * H_DIM + k];
    }
    __syncthreads();

    const int lane = t & 31, wave = t >> 5;
    const int n0 = wave * 16;
    const int mm = lane & 15;
    const int hi = lane >> 4;
    const int kb = hi * 2;
    const float rd = sRD[mm];

    v8f c = {0.f, 0.f, 0.f, 0.f, 0.f, 0.f, 0.f, 0.f};
    for (int k = 0; k < H_DIM; k += 4) {             // (agg/deg) @ Wl
        v2f a, b;
        a.x = sA[mm * H_DIM + k + kb]     * rd;
        a.y = sA[mm * H_DIM + k + kb + 1] * rd;
        b.x = Wl[(k + kb)     * H_DIM + n0 + mm];
        b.y = Wl[(k + kb + 1) * H_DIM + n0 + mm];
        c = __builtin_amdgcn_wmma_f32_16x16x4_f32(false, a, false, b,
                                                  (short)0, c, false, false);
    }
    for (int k = 0; k < H_DIM; k += 4) {             // + x_src @ Wr
        v2f a, b;
        a.x = sX[mm * H_DIM + k + kb];
        a.y = sX[mm * H_DIM + k + kb + 1];
        b.x = Wr[(k + kb)     * H_DIM + n0 + mm];
        b.y = Wr[(k + kb + 1) * H_DIM + n0 + mm];
        c = __builtin_amdgcn_wmma_f32_16x16x4_f32(false, a, false, b,
                                                  (short)0, c, false, false);
    }

    const float bv = bl[n0 + mm];
    float vals[8];
#pragma unroll
    for (int r = 0; r < 8; ++r) vals[r] = c[r] + bv;

    // Partial row sum-of-squares: reduce across the 16 lanes of each half-wave.
#pragma unroll
    for (int r = 0; r < 8; ++r) {
        float s = vals[r] * vals[r];
        s += __shfl_xor(s, 1);
        s += __shfl_xor(s, 2);
        s += __shfl_xor(s, 4);
        s += __shfl_xor(s, 8);
        if (mm == 0) sPart[wave][r + 8 * hi] = s;    // lane 0 -> rows 0..7, lane 16 -> rows 8..15
    }
    __syncthreads();

    const int   col = n0 + mm;
    const float bscale = bng[col] / sqrtf(bnv[col] + 1e-5f);
    const float bshift = bnb[col] - bnm[col] * bscale;
#pragma unroll
    for (int r = 0; r < 8; ++r) {
        const int row = r + 8 * hi;
        float tot = 0.f;
#pragma unroll
        for (int w = 0; w < 8; ++w) tot += sPart[w][row];
        const float rn = 1.0f / fmaxf(sqrtf(tot), 1e-12f);
        float y = vals[r] * rn + hold[(size_t)(m0 + row) * H_DIM + col];
        y = y * bscale + bshift;
        y = fmaxf(y, 0.f);
        hnew[(size_t)(m0 + row) * H_DIM + col] = y;
    }
}

// ---------------------------------------------------------------------------
// Host orchestration
// ---------------------------------------------------------------------------
extern "C" void kernel_launch(void* const* d_in, const int* in_sizes, int n_in,
                              void* d_out, int out_size, void* d_ws, size_t ws_size,
                              hipStream_t stream) {
    const float* x_user     = (const float*)d_in[0];
    const float* x_item     = (const float*)d_in[1];
    const float* enc_W_user = (const float*)d_in[2];
    const float* enc_b_user = (const float*)d_in[3];
    const float* enc_W_item = (const float*)d_in[4];
    const float* enc_b_item = (const float*)d_in[5];
    const float* Wl_ui      = (const float*)d_in[6];
    const float* bl_ui      = (const float*)d_in[7];
    const float* Wr_ui      = (const float*)d_in[8];
    const float* Wl_iu      = (const float*)d_in[9];
    const float* bl_iu      = (const float*)d_in[10];
    const float* Wr_iu      = (const float*)d_in[11];
    const float* bn_g_user  = (const float*)d_in[12];
    const float* bn_b_user  = (const float*)d_in[13];
    const float* bn_m_user  = (const float*)d_in[14];
    const float* bn_v_user  = (const float*)d_in[15];
    const float* bn_g_item  = (const float*)d_in[16];
    const float* bn_b_item  = (const float*)d_in[17];
    const float* bn_m_item  = (const float*)d_in[18];
    const float* bn_v_item  = (const float*)d_in[19];
    const float* cls_W1     = (const float*)d_in[20];
    const float* cls_b1     = (const float*)d_in[21];
    const float* cls_W2     = (const float*)d_in[22];
    const float* cls_b2     = (const float*)d_in[23];
    const int*   ei_ui      = (const int*)d_in[24];   // [2,E]: row0 src, row1 dst
    const int*   ei_iu      = (const int*)d_in[25];

    const size_t NH = (size_t)N_NODES * H_DIM;
    float* ws     = (float*)d_ws;
    float* hu_a   = ws;
    float* hu_b   = ws + NH;
    float* hi_a   = ws + 2 * NH;
    float* hi_b   = ws + 3 * NH;
    float* agg    = ws + 4 * NH;          // reused as classifier hidden at the end
    float* deg_ui = ws + 5 * NH;
    float* deg_iu = deg_ui + N_NODES;

    const int rowTiles = N_NODES / 16;    // 6250
    const dim3 g1(rowTiles);

    // Encoders: h = relu(x @ W + b)
    gemm_bias_act<<<g1, 256, 0, stream>>>(x_user, enc_W_user, enc_b_user, hu_a, 128, 128, 1);
    gemm_bias_act<<<g1, 256, 0, stream>>>(x_item, enc_W_item, enc_b_item, hi_a, 128, 128, 1);

    // Degrees (layer-invariant)
    hipMemsetAsync(deg_ui, 0, 2 * (size_t)N_NODES * sizeof(float), stream);
    degree_kernel<<<(E_EDGES + 255) / 256, 256, 0, stream>>>(ei_ui + E_EDGES, E_EDGES, deg_ui);
    degree_kernel<<<(E_EDGES + 255) / 256, 256, 0, stream>>>(ei_iu + E_EDGES, E_EDGES, deg_iu);

    float* hu_cur = hu_a; float* hu_nxt = hu_b;
    float* hi_cur = hi_a; float* hi_nxt = hi_b;
    const dim3 gE(E_EDGES / 8);           // 8 edges (waves) per 256-thread block

    for (int l = 0; l < N_LAYERS; ++l) {
        const size_t oW = (size_t)l * H_DIM * H_DIM;
        const size_t oB = (size_t)l * H_DIM;

        // m_item = SAGE(h_u via user->item); h_i' = relu(BN_item(m_item + h_i))
        hipMemsetAsync(agg, 0, NH * sizeof(float), stream);
        scatter_add_kernel<<<gE, 256, 0, stream>>>(hu_cur, ei_ui, ei_ui + E_EDGES, E_EDGES, agg);
        sage_fused_kernel<<<g1, 256, 0, stream>>>(agg, deg_ui, hu_cur, hi_cur,
            Wl_ui + oW, bl_ui + oB, Wr_ui + oW,
            bn_g_item + oB, bn_b_item + oB, bn_m_item + oB, bn_v_item + oB, hi_nxt);

        // m_user = SAGE(h_i via item->user); h_u' = relu(BN_user(m_user + h_u))
        hipMemsetAsync(agg, 0, NH * sizeof(float), stream);
        scatter_add_kernel<<<gE, 256, 0, stream>>>(hi_cur, ei_iu, ei_iu + E_EDGES, E_EDGES, agg);
        sage_fused_kernel<<<g1, 256, 0, stream>>>(agg, deg_iu, hi_cur, hu_cur,
            Wl_iu + oW, bl_iu + oB, Wr_iu + oW,
            bn_g_user + oB, bn_b_user + oB, bn_m_user + oB, bn_v_user + oB, hu_nxt);

        float* t0 = hu_cur; hu_cur = hu_nxt; hu_nxt = t0;
        float* t1 = hi_cur; hi_cur = hi_nxt; hi_nxt = t1;
    }

    // Classifier head on user nodes: relu(h_u @ W1 + b1) @ W2 + b2
    float* hid = agg;                     // [N, 64], agg buffer is free now
    gemm_bias_act<<<g1, 128, 0, stream>>>(hu_cur, cls_W1, cls_b1, hid, 128, 64, 1);
    gemm_bias_act<<<g1,  64, 0, stream>>>(hid, cls_W2, cls_b2, (float*)d_out, 64, 32, 0);
}